// AttentionCell_25606595019318
// MI455X (gfx1250) — compile-verified
//
#include <hip/hip_runtime.h>
#include <hip/hip_bf16.h>

typedef __attribute__((ext_vector_type(2))) float v2f;
typedef __attribute__((ext_vector_type(8))) float v8f;

#define BB 8
#define TT 2048
#define VV 256
#define DD 512
#define TILE 16
#define NTILES (TT / TILE)   // 128

// One workgroup handles one (b, 16-timestep tile).
//   Phase 0: stage the 32 clamped history symbol ids (positions t0-15 .. t0+16).
//   Phase 1: scores S[r,q] = <enc[t0+r,:], M[s_q,:]> via f32 WMMA.
//            All 4 waves: wave w -> q-tile (w&1), K-half (w>>1); partials in LDS.
//            Dual accumulators per wave to break the WMMA RAW chain.
//   Phase 2: reduce K-halves + per-row softmax over the 16-wide diagonal band
//            -> banded P[16][32].
//   Phase 3: compressed = P (16x32) * Crows (32x512) via f32 WMMA (dual acc).
//   Phase 4: copy encodings into out[..., 512:1024] (coalesced float4).
__global__ __launch_bounds__(128) void attention_cell_kernel(
    const int* __restrict__ symbols,     // (B, T)
    const float* __restrict__ enc,       // (B, T, D)
    const float* __restrict__ M,         // (V, D)
    const float* __restrict__ C,         // (V, D)
    float* __restrict__ out,             // (B, T, 2*D)
    float* __restrict__ pout)            // (B, T, 16)
{
    __shared__ int   s_sym[32];
    __shared__ float s_S[2][16][33];  // [K-half][row][q], stride-33 vs bank conflicts
    __shared__ float s_P[16][33];     // banded softmax matrix

    const int blk  = blockIdx.x;            // 0 .. B*NTILES-1
    const int b    = blk / NTILES;
    const int t0   = (blk % NTILES) * TILE;
    const int tid  = threadIdx.x;
    const int wave = tid >> 5;
    const int lane = tid & 31;

    // ---- Phase 0: history symbols (clamped; pos<0 pads with symbols[b,0]) ----
    if (tid < 32) {
        int pos = t0 - 15 + tid;
        pos = pos < 0 ? 0 : pos;
        pos = pos > (TT - 1) ? (TT - 1) : pos;
        s_sym[tid] = symbols[b * TT + pos];
    }
    __syncthreads();

    const int m16 = lane & 15;              // row (A) / col (B) index
    const int hk  = (lane >> 4) << 1;       // K sub-offset: 0 or 2

    // ---- Phase 1: scores GEMM, all 4 waves ----
    {
        const int qt   = wave & 1;                     // q tile (cols 0-15 / 16-31)
        const int part = wave >> 1;                    // K half: [0,256) or [256,512)
        const int kbeg = part * (DD / 2);

        const float* encrow = enc + ((size_t)(b * TT + t0 + m16)) * DD + hk + kbeg;
        const float* Mrow   = M   + ((size_t)s_sym[qt * 16 + m16]) * DD + hk + kbeg;

        v8f acc0 = {}, acc1 = {};
        for (int k = 0; k < DD / 2; k += 8) {
            v2f a0 = *(const v2f*)(encrow + k);
            v2f b0 = *(const v2f*)(Mrow   + k);
            acc0 = __builtin_amdgcn_wmma_f32_16x16x4_f32(
                false, a0, false, b0, (short)0, acc0, false, false);
            v2f a1 = *(const v2f*)(encrow + k + 4);
            v2f b1 = *(const v2f*)(Mrow   + k + 4);
            acc1 = __builtin_amdgcn_wmma_f32_16x16x4_f32(
                false, a1, false, b1, (short)0, acc1, false, false);
        }
        v8f acc = acc0 + acc1;

        const int row0 = (lane >> 4) << 3;             // 0 or 8
        const int col  = qt * 16 + m16;
        #pragma unroll
        for (int i = 0; i < 8; ++i)
            s_S[part][row0 + i][col] = acc[i];
    }
    __syncthreads();

    // ---- Phase 2: reduce halves + softmax over the band of each row ----
    if (tid < 16) {
        const int r = tid;
        float vals[16];
        float mx = -3.0e38f;
        #pragma unroll
        for (int l = 0; l < 16; ++l) {
            float v = s_S[0][r][r + l] + s_S[1][r][r + l];  // q = r + l
            vals[l] = v;
            mx = fmaxf(mx, v);
        }
        float sum = 0.0f;
        #pragma unroll
        for (int l = 0; l < 16; ++l) {
            float e = expf(vals[l] - mx);
            vals[l] = e;
            sum += e;
        }
        const float inv = 1.0f / sum;
        #pragma unroll
        for (int q = 0; q < 32; ++q) s_P[r][q] = 0.0f;  // zero outside the band
        float* prow = pout + ((size_t)(b * TT + t0 + r)) * 16;
        #pragma unroll
        for (int l = 0; l < 16; ++l) {
            float p = vals[l] * inv;
            s_P[r][r + l] = p;
            prow[l] = p;
        }
    }
    __syncthreads();

    // ---- Phase 3: compressed = P(16x32) * C_rows(32x512), WMMA over 32 d-tiles ----
    for (int nt = wave; nt < DD / 16; nt += 4) {
        const int dcol = nt * 16 + m16;
        v8f acc0 = {}, acc1 = {};
        #pragma unroll
        for (int kc = 0; kc < 16; kc += 4) {
            {
                const int k = kc + hk;
                v2f a; a.x = s_P[m16][k]; a.y = s_P[m16][k + 1];
                v2f bc;
                bc.x = C[(size_t)s_sym[k]     * DD + dcol];
                bc.y = C[(size_t)s_sym[k + 1] * DD + dcol];
                acc0 = __builtin_amdgcn_wmma_f32_16x16x4_f32(
                    false, a, false, bc, (short)0, acc0, false, false);
            }
            {
                const int k = kc + 16 + hk;
                v2f a; a.x = s_P[m16][k]; a.y = s_P[m16][k + 1];
                v2f bc;
                bc.x = C[(size_t)s_sym[k]     * DD + dcol];
                bc.y = C[(size_t)s_sym[k + 1] * DD + dcol];
                acc1 = __builtin_amdgcn_wmma_f32_16x16x4_f32(
                    false, a, false, bc, (short)0, acc1, false, false);
            }
        }
        v8f acc = acc0 + acc1;

        const int row0 = (lane >> 4) << 3;
        #pragma unroll
        for (int i = 0; i < 8; ++i)
            out[((size_t)(b * TT + t0 + row0 + i)) * (2 * DD) + dcol] = acc[i];
    }

    // ---- Phase 4: out[..., 512:1024] = encodings (coalesced float4 copy) ----
    for (int idx = tid; idx < TILE * (DD / 4); idx += 128) {
        const int r  = idx >> 7;          // / (512/4)
        const int c4 = idx & 127;
        const float4* src = (const float4*)(enc + ((size_t)(b * TT + t0 + r)) * DD) + c4;
        float4*       dst = (float4*)(out + ((size_t)(b * TT + t0 + r)) * (2 * DD) + DD) + c4;
        *dst = *src;
    }
}

extern "C" void kernel_launch(void* const* d_in, const int* in_sizes, int n_in,
                              void* d_out, int out_size, void* d_ws, size_t ws_size,
                              hipStream_t stream) {
    const int*   symbols = (const int*)d_in[0];
    const float* enc     = (const float*)d_in[1];
    const float* M       = (const float*)d_in[2];
    const float* C       = (const float*)d_in[3];

    float* out  = (float*)d_out;
    float* pout = out + (size_t)BB * TT * (2 * DD);   // p follows output in d_out

    dim3 grid(BB * NTILES);   // 1024 workgroups
    dim3 block(128);          // 4 wave32
    attention_cell_kernel<<<grid, block, 0, stream>>>(symbols, enc, M, C, out, pout);
}